// ProposedGCN_4569845203117
// MI455X (gfx1250) — compile-verified
//
#include <hip/hip_runtime.h>

#define N_NODES 50000
#define N_EDGES 1600000
#define IN_DIM  128
#define HID_DIM 64
#define OUT_DIM 3

typedef __attribute__((ext_vector_type(2))) float v2f;
typedef __attribute__((ext_vector_type(8))) float v8f;

// ---------------------------------------------------------------------------
// Init: deg = 1 (self-loop), agg1 = 0, agg2 = 0
// ---------------------------------------------------------------------------
__global__ __launch_bounds__(256) void gcn_init_kernel(float* __restrict__ agg1,
                                                       float* __restrict__ agg2,
                                                       float* __restrict__ deg) {
    size_t idx = (size_t)blockIdx.x * 256 + threadIdx.x;
    if (idx < (size_t)N_NODES * HID_DIM) agg1[idx] = 0.0f;
    if (idx < (size_t)N_NODES * OUT_DIM) agg2[idx] = 0.0f;
    if (idx < (size_t)N_NODES)           deg[idx]  = 1.0f;  // self loop
}

// ---------------------------------------------------------------------------
// In-degree via f32 atomics (edge_index is int64 per reference)
// ---------------------------------------------------------------------------
__global__ __launch_bounds__(256) void gcn_deg_kernel(const long long* __restrict__ ei,
                                                      float* __restrict__ deg) {
    int e = blockIdx.x * 256 + threadIdx.x;
    if (e < N_EDGES) {
        long long d = ei[N_EDGES + e];
        atomicAdd(&deg[d], 1.0f);
    }
}

__global__ __launch_bounds__(256) void gcn_dinv_kernel(float* __restrict__ deg) {
    int i = blockIdx.x * 256 + threadIdx.x;
    if (i < N_NODES) {
        float d = deg[i];
        deg[i] = (d > 0.0f) ? rsqrtf(d) : 0.0f;   // becomes dinv in place
    }
}

// ---------------------------------------------------------------------------
// lin1: t1[50000x64] = x[50000x128] @ W1[128x64]  (f32 WMMA 16x16x4)
// One wave per 32-row strip (2 M-tiles share B fragments). W1 staged in LDS
// pair-interleaved: sW[(kpair*64 + col)*2 + (k&1)] so each B fragment is one
// aligned ds_load_b64 into an even VGPR pair (no repack movs).
// ---------------------------------------------------------------------------
__global__ __launch_bounds__(256) void gcn_lin1_wmma_kernel(const float* __restrict__ x,
                                                            const float* __restrict__ W1,
                                                            float* __restrict__ t1) {
    __shared__ float sW[IN_DIM * HID_DIM];   // 32 KB, pair-interleaved layout
    for (int i = threadIdx.x; i < IN_DIM * HID_DIM; i += 256) {
        int k = i >> 6;            // W1 row (K index)
        int c = i & 63;            // W1 col (N index)
        sW[((k >> 1) * HID_DIM + c) * 2 + (k & 1)] = W1[i];
    }
    __syncthreads();

    const int lane  = threadIdx.x & 31;
    const int wave  = threadIdx.x >> 5;
    const int strip = blockIdx.x * 8 + wave;          // 32-row strip index
    const int mbase = strip * 32;
    if (mbase >= N_NODES) return;                     // wave-uniform: EXEC stays full
    const int half  = lane >> 4;                      // 0: lanes 0-15, 1: lanes 16-31
    const int l16   = lane & 15;
    const bool hasT1 = (mbase + 32 <= N_NODES);       // wave-uniform tail guard

    int row0 = mbase + l16;
    int row1 = mbase + 16 + l16;
    if (row1 >= N_NODES) row1 = N_NODES - 1;          // clamp (loads only; stores guarded)

    const float* __restrict__ arow0 = x + (size_t)row0 * IN_DIM;
    const float* __restrict__ arow1 = x + (size_t)row1 * IN_DIM;
    const v2f*   __restrict__ sWp   = (const v2f*)sW; // pair-interleaved pairs

    v8f acc[2][4];
    for (int t = 0; t < 2; ++t)
        for (int n = 0; n < 4; ++n)
            acc[t][n] = (v8f){};

    for (int k0 = 0; k0 < IN_DIM; k0 += 4) {
        // A fragments (16x4 f32): lanes 0-15 K={k0,k0+1}; lanes 16-31 K={k0+2,k0+3}
        v2f a0 = *(const v2f*)(arow0 + k0 + 2 * half);
        v2f a1 = *(const v2f*)(arow1 + k0 + 2 * half);
        // B fragments (4x16 f32): single b64 LDS load per n-tile
        const int kp = (k0 >> 1) + half;              // K-pair index for this lane half
        v2f b0 = sWp[kp * HID_DIM +  0 + l16];
        v2f b1 = sWp[kp * HID_DIM + 16 + l16];
        v2f b2 = sWp[kp * HID_DIM + 32 + l16];
        v2f b3 = sWp[kp * HID_DIM + 48 + l16];

        acc[0][0] = __builtin_amdgcn_wmma_f32_16x16x4_f32(false, a0, false, b0, (short)0, acc[0][0], false, false);
        acc[0][1] = __builtin_amdgcn_wmma_f32_16x16x4_f32(false, a0, false, b1, (short)0, acc[0][1], false, false);
        acc[0][2] = __builtin_amdgcn_wmma_f32_16x16x4_f32(false, a0, false, b2, (short)0, acc[0][2], false, false);
        acc[0][3] = __builtin_amdgcn_wmma_f32_16x16x4_f32(false, a0, false, b3, (short)0, acc[0][3], false, false);
        acc[1][0] = __builtin_amdgcn_wmma_f32_16x16x4_f32(false, a1, false, b0, (short)0, acc[1][0], false, false);
        acc[1][1] = __builtin_amdgcn_wmma_f32_16x16x4_f32(false, a1, false, b1, (short)0, acc[1][1], false, false);
        acc[1][2] = __builtin_amdgcn_wmma_f32_16x16x4_f32(false, a1, false, b2, (short)0, acc[1][2], false, false);
        acc[1][3] = __builtin_amdgcn_wmma_f32_16x16x4_f32(false, a1, false, b3, (short)0, acc[1][3], false, false);
    }

    // C/D layout: VGPR r -> M = tilebase + r + 8*half, N = ntile*16 + l16
    for (int t = 0; t < 2; ++t) {
        if (t == 1 && !hasT1) break;                  // wave-uniform
        const int tb = mbase + t * 16 + half * 8;
        for (int r = 0; r < 8; ++r) {
            float* __restrict__ orow = t1 + (size_t)(tb + r) * HID_DIM;
            orow[ 0 + l16] = acc[t][0][r];
            orow[16 + l16] = acc[t][1][r];
            orow[32 + l16] = acc[t][2][r];
            orow[48 + l16] = acc[t][3][r];
        }
    }
}

// ---------------------------------------------------------------------------
// Layer-1 edge aggregation: 64 threads per edge (feature-coalesced), L2 atomics
// ---------------------------------------------------------------------------
__global__ __launch_bounds__(256) void gcn_agg1_kernel(const long long* __restrict__ ei,
                                                       const float* __restrict__ t1,
                                                       const float* __restrict__ dinv,
                                                       float* __restrict__ agg1) {
    size_t idx = (size_t)blockIdx.x * 256 + threadIdx.x;
    if (idx >= (size_t)N_EDGES * HID_DIM) return;
    int e = (int)(idx >> 6);
    int f = (int)(idx & 63);
    long long s = ei[e];
    long long d = ei[N_EDGES + e];
    float nrm = dinv[s] * dinv[d];
    atomicAdd(&agg1[(size_t)d * HID_DIM + f], t1[(size_t)s * HID_DIM + f] * nrm);
}

// finalize1: add self-loop term + bias, ReLU; result written in place (agg1 -> h)
__global__ __launch_bounds__(256) void gcn_fin1_kernel(float* __restrict__ agg1,
                                                       const float* __restrict__ t1,
                                                       const float* __restrict__ dinv,
                                                       const float* __restrict__ b1) {
    size_t idx = (size_t)blockIdx.x * 256 + threadIdx.x;
    if (idx >= (size_t)N_NODES * HID_DIM) return;
    int i = (int)(idx >> 6);
    int f = (int)(idx & 63);
    float di = dinv[i];
    float v  = agg1[idx] + t1[idx] * di * di + b1[f];
    agg1[idx] = fmaxf(v, 0.0f);
}

// ---------------------------------------------------------------------------
// lin2: t2[50000x3] = h[50000x64] @ W2[64x3]  (f32 WMMA, N padded to 16)
// Same pair-interleaved LDS trick for the B operand.
// ---------------------------------------------------------------------------
__global__ __launch_bounds__(256) void gcn_lin2_wmma_kernel(const float* __restrict__ h,
                                                            const float* __restrict__ W2,
                                                            float* __restrict__ t2) {
    __shared__ float sW[HID_DIM * 16];               // pair-interleaved, zero-padded
    for (int i = threadIdx.x; i < HID_DIM * 16; i += 256) {
        int k = i >> 4, c = i & 15;
        float v = (c < OUT_DIM) ? W2[k * OUT_DIM + c] : 0.0f;
        sW[((k >> 1) * 16 + c) * 2 + (k & 1)] = v;
    }
    __syncthreads();

    const int lane  = threadIdx.x & 31;
    const int wave  = threadIdx.x >> 5;
    const int strip = blockIdx.x * 8 + wave;
    if (strip * 16 >= N_NODES) return;                // wave-uniform
    const int mbase = strip * 16;
    const int half  = lane >> 4;
    const int l16   = lane & 15;

    const float* __restrict__ arow = h + (size_t)(mbase + l16) * HID_DIM;
    const v2f*   __restrict__ sWp  = (const v2f*)sW;

    v8f acc = {};
    for (int k0 = 0; k0 < HID_DIM; k0 += 4) {
        v2f a = *(const v2f*)(arow + k0 + 2 * half);
        v2f b = sWp[((k0 >> 1) + half) * 16 + l16];
        acc = __builtin_amdgcn_wmma_f32_16x16x4_f32(false, a, false, b, (short)0, acc, false, false);
    }

    if (l16 < OUT_DIM) {
        for (int r = 0; r < 8; ++r)
            t2[(size_t)(mbase + half * 8 + r) * OUT_DIM + l16] = acc[r];
    }
}

// Layer-2 edge aggregation: 4 threads per edge (3 active features)
__global__ __launch_bounds__(256) void gcn_agg2_kernel(const long long* __restrict__ ei,
                                                       const float* __restrict__ t2,
                                                       const float* __restrict__ dinv,
                                                       float* __restrict__ agg2) {
    size_t idx = (size_t)blockIdx.x * 256 + threadIdx.x;
    if (idx >= (size_t)N_EDGES * 4) return;
    int e = (int)(idx >> 2);
    int f = (int)(idx & 3);
    long long s = ei[e];
    long long d = ei[N_EDGES + e];
    float nrm = dinv[s] * dinv[d];
    if (f < OUT_DIM)
        atomicAdd(&agg2[(size_t)d * OUT_DIM + f], t2[(size_t)s * OUT_DIM + f] * nrm);
}

// finalize2: self-loop + bias, then log_softmax over 3 classes
__global__ __launch_bounds__(256) void gcn_fin2_kernel(const float* __restrict__ agg2,
                                                       const float* __restrict__ t2,
                                                       const float* __restrict__ dinv,
                                                       const float* __restrict__ b2,
                                                       float* __restrict__ out) {
    int i = blockIdx.x * 256 + threadIdx.x;
    if (i >= N_NODES) return;
    float di2 = dinv[i] * dinv[i];
    float o0 = agg2[(size_t)i * 3 + 0] + t2[(size_t)i * 3 + 0] * di2 + b2[0];
    float o1 = agg2[(size_t)i * 3 + 1] + t2[(size_t)i * 3 + 1] * di2 + b2[1];
    float o2 = agg2[(size_t)i * 3 + 2] + t2[(size_t)i * 3 + 2] * di2 + b2[2];
    float m   = fmaxf(o0, fmaxf(o1, o2));
    float lse = m + logf(expf(o0 - m) + expf(o1 - m) + expf(o2 - m));
    out[(size_t)i * 3 + 0] = o0 - lse;
    out[(size_t)i * 3 + 1] = o1 - lse;
    out[(size_t)i * 3 + 2] = o2 - lse;
}

// ---------------------------------------------------------------------------
extern "C" void kernel_launch(void* const* d_in, const int* in_sizes, int n_in,
                              void* d_out, int out_size, void* d_ws, size_t ws_size,
                              hipStream_t stream) {
    const float*     x  = (const float*)d_in[0];
    const long long* ei = (const long long*)d_in[1];   // int64 edge_index [2, E]
    const float*     W1 = (const float*)d_in[2];
    const float*     b1 = (const float*)d_in[3];
    const float*     W2 = (const float*)d_in[4];
    const float*     b2 = (const float*)d_in[5];
    float*           out = (float*)d_out;

    // Workspace layout (floats), ~27 MB total
    float* ws   = (float*)d_ws;
    size_t off  = 0;
    float* dinv = ws + off; off += 50048;                       // deg -> dinv (in place)
    float* t1   = ws + off; off += (size_t)N_NODES * HID_DIM;   // 3.2M
    float* agg1 = ws + off; off += (size_t)N_NODES * HID_DIM;   // 3.2M (becomes h)
    float* t2   = ws + off; off += 150016;
    float* agg2 = ws + off; off += 150016;
    (void)ws_size; (void)in_sizes; (void)n_in; (void)out_size;

    const int B = 256;
    const int gInit  = (N_NODES * HID_DIM + B - 1) / B;          // 12500
    const int gEdge  = (N_EDGES + B - 1) / B;                    // 6250
    const int gNode  = (N_NODES + B - 1) / B;                    // 196
    const int gGemm1 = (((N_NODES + 31) / 32) + 7) / 8;          // 196 (1563 strips of 32)
    const int gGemm2 = ((N_NODES / 16) + 7) / 8;                 // 391 (3125 strips of 16)
    const int gAgg1  = (int)(((size_t)N_EDGES * HID_DIM + B - 1) / B);  // 400000
    const int gAgg2  = (int)(((size_t)N_EDGES * 4 + B - 1) / B);        // 25000

    gcn_init_kernel<<<gInit, B, 0, stream>>>(agg1, agg2, dinv);
    gcn_deg_kernel<<<gEdge, B, 0, stream>>>(ei, dinv);
    gcn_dinv_kernel<<<gNode, B, 0, stream>>>(dinv);

    gcn_lin1_wmma_kernel<<<gGemm1, B, 0, stream>>>(x, W1, t1);
    gcn_agg1_kernel<<<gAgg1, B, 0, stream>>>(ei, t1, dinv, agg1);
    gcn_fin1_kernel<<<gInit, B, 0, stream>>>(agg1, t1, dinv, b1);   // agg1 -> h

    gcn_lin2_wmma_kernel<<<gGemm2, B, 0, stream>>>(agg1, W2, t2);
    gcn_agg2_kernel<<<gAgg2, B, 0, stream>>>(ei, t2, dinv, agg2);
    gcn_fin2_kernel<<<gNode, B, 0, stream>>>(agg2, t2, dinv, b2, out);
}